// GraphSage_15668040696564
// MI455X (gfx1250) — compile-verified
//
#include <hip/hip_runtime.h>
#include <math.h>

#define F_IN 64
#define DIM  128
#define SNB  16
#define MT   32    // nodes per block (two 16-row M-tiles)

typedef __attribute__((ext_vector_type(2))) float v2f;
typedef __attribute__((ext_vector_type(4))) float v4f;
typedef __attribute__((ext_vector_type(8))) float v8f;

__device__ __forceinline__ v8f wmma_f32(v2f a, v2f b, v8f c) {
    return __builtin_amdgcn_wmma_f32_16x16x4_f32(
        false, a, false, b, (short)0, c, false, false);
}

// ---------------------------------------------------------------------------
// Kernel 1: x = features @ w_feat.T
// Block = 32 nodes, 256 threads (8 waves). Wave nt owns output cols nt*16..+15
// and runs two independent WMMA accumulator chains (rows 0-15 and 16-31)
// sharing each B fragment.
// ---------------------------------------------------------------------------
__global__ __launch_bounds__(256) void sage_feat_kernel(
    const float* __restrict__ feat, const float* __restrict__ wfeat,
    float* __restrict__ xout, int N)
{
    __shared__ float fS[MT][F_IN + 4];   // 272B stride: conflict-free, 16B aligned
    __shared__ float xS[MT][DIM + 4];    // 528B stride: conflict-free, 16B aligned

    const int t  = threadIdx.x;
    const int n0 = blockIdx.x * MT;

    // Stage the 32x64 A tile into LDS (coalesced float4 per thread, 2 rows each)
    #pragma unroll
    for (int mm = (t >> 4); mm < MT; mm += 16) {
        int c = (t & 15) * 4;
        int row = n0 + mm; if (row >= N) row = N - 1;
        *(v4f*)&fS[mm][c] = *(const v4f*)(feat + (size_t)row * F_IN + c);
    }
    __syncthreads();

    const int wave = t >> 5;
    const int lane = t & 31;
    const int lm   = lane & 15;      // A row within tile / B col within tile
    const int lh   = lane >> 4;      // K half-select
    const int ncol = wave * 16 + lm;
    const float* wrow = wfeat + (size_t)ncol * F_IN;   // B[k][ncol] = wfeat[ncol][k]

    v8f acc0 = {}, acc1 = {};
    #pragma unroll
    for (int k0 = 0; k0 < F_IN; k0 += 4) {
        v2f b  = *(const v2f*)(wrow + k0 + lh * 2);
        v2f a0 = *(const v2f*)&fS[lm][k0 + lh * 2];
        v2f a1 = *(const v2f*)&fS[16 + lm][k0 + lh * 2];
        acc0 = wmma_f32(a0, b, acc0);    // two independent chains: no RAW stall
        acc1 = wmma_f32(a1, b, acc1);
    }

    // D layout: VGPR r -> row M = lh*8 + r, column ncol. Stage to LDS.
    #pragma unroll
    for (int r = 0; r < 8; ++r) {
        xS[lh * 8 + r][ncol]      = acc0[r];
        xS[16 + lh * 8 + r][ncol] = acc1[r];
    }
    __syncthreads();

    // Coalesced b128 stores (512B per 16 threads)
    #pragma unroll
    for (int mm = (t >> 4); mm < MT; mm += 16) {
        int c = (t & 15) * 8;
        int row = n0 + mm;
        if (row < N) {
            *(v4f*)(xout + (size_t)row * DIM + c)     = *(const v4f*)&xS[mm][c];
            *(v4f*)(xout + (size_t)row * DIM + c + 4) = *(const v4f*)&xS[mm][c + 4];
        }
    }
}

// ---------------------------------------------------------------------------
// Kernel 2: one SAGE layer, fused: gather-mean -> GEMM(WMMA) -> ReLU -> L2norm
// Block = 32 nodes, 256 threads (8 waves). Each block reads the 64KB weight
// matrix from L2 once for 32 nodes (half the traffic of a 16-node block).
// ---------------------------------------------------------------------------
__global__ __launch_bounds__(256) void sage_layer_kernel(
    const float* __restrict__ x, const int* __restrict__ neigh,
    const float* __restrict__ w, float* __restrict__ out, int N)
{
    __shared__ float aggS[MT][DIM + 4];
    __shared__ float hS[MT][DIM + 4];
    __shared__ int   sidx[MT][SNB];
    __shared__ float rnorm[MT];

    const int t  = threadIdx.x;
    const int n0 = blockIdx.x * MT;

    // Load neighbor index tile (32 nodes x 16 ids, coalesced)
    #pragma unroll
    for (int i = t; i < MT * SNB; i += 256) {
        int m = i >> 4, s = i & 15;
        int row = n0 + m; if (row >= N) row = N - 1;
        sidx[m][s] = neigh[(size_t)row * SNB + s];
    }
    if (t < MT) rnorm[t] = 0.0f;
    __syncthreads();

    // Gather + mean over {self} U neighbors. Thread (m, cg=t&15) owns 8 cols;
    // a wave touches only 2 random source rows per iteration and consumes each
    // 512B row fully (L2-friendly).
    #pragma unroll
    for (int mm = (t >> 4); mm < MT; mm += 16) {
        int c = (t & 15) * 8;
        int self = n0 + mm; if (self >= N) self = N - 1;
        const float* xr = x + (size_t)self * DIM + c;
        v4f a0 = *(const v4f*)(xr);
        v4f a1 = *(const v4f*)(xr + 4);
        #pragma unroll 4
        for (int s = 0; s < SNB; ++s) {
            int r = sidx[mm][s];
            const float* xs = x + (size_t)r * DIM + c;
            a0 += *(const v4f*)(xs);
            a1 += *(const v4f*)(xs + 4);
        }
        const float inv = 1.0f / (float)(SNB + 1);
        a0 *= inv; a1 *= inv;
        *(v4f*)&aggS[mm][c]     = a0;
        *(v4f*)&aggS[mm][c + 4] = a1;
    }
    __syncthreads();

    // WMMA: h = agg @ w.T ; wave nt owns cols nt*16..+15, K = 128, two M-tiles.
    const int wave = t >> 5;
    const int lane = t & 31;
    const int lm   = lane & 15;
    const int lh   = lane >> 4;
    const int ncol = wave * 16 + lm;
    const float* wrow = w + (size_t)ncol * DIM;   // B[k][ncol] = w[ncol][k]

    v8f acc0 = {}, acc1 = {};
    #pragma unroll
    for (int k0 = 0; k0 < DIM; k0 += 4) {
        v2f b  = *(const v2f*)(wrow + k0 + lh * 2);
        v2f a0 = *(const v2f*)&aggS[lm][k0 + lh * 2];
        v2f a1 = *(const v2f*)&aggS[16 + lm][k0 + lh * 2];
        acc0 = wmma_f32(a0, b, acc0);
        acc1 = wmma_f32(a1, b, acc1);
    }

    // ReLU, stash D tiles to LDS for the cross-wave row reduction.
    #pragma unroll
    for (int r = 0; r < 8; ++r) {
        float v0 = acc0[r] > 0.0f ? acc0[r] : 0.0f;
        float v1 = acc1[r] > 0.0f ? acc1[r] : 0.0f;
        hS[lh * 8 + r][ncol]      = v0;
        hS[16 + lh * 8 + r][ncol] = v1;
    }
    __syncthreads();

    // Row sum-of-squares: 16 threads per row, 8 elements each, ds_add_f32.
    #pragma unroll
    for (int mm = (t >> 4); mm < MT; mm += 16) {
        int c = (t & 15) * 8;
        v4f h0 = *(const v4f*)&hS[mm][c];
        v4f h1 = *(const v4f*)&hS[mm][c + 4];
        float ss = h0.x*h0.x + h0.y*h0.y + h0.z*h0.z + h0.w*h0.w
                 + h1.x*h1.x + h1.y*h1.y + h1.z*h1.z + h1.w*h1.w;
        atomicAdd(&rnorm[mm], ss);
    }
    __syncthreads();

    // Normalize and store (coalesced 512B per 16 threads).
    #pragma unroll
    for (int mm = (t >> 4); mm < MT; mm += 16) {
        int c = (t & 15) * 8;
        int row = n0 + mm;
        if (row < N) {
            float inv = 1.0f / fmaxf(sqrtf(rnorm[mm]), 1e-12f);
            v4f h0 = *(const v4f*)&hS[mm][c]     * inv;
            v4f h1 = *(const v4f*)&hS[mm][c + 4] * inv;
            *(v4f*)(out + (size_t)row * DIM + c)     = h0;
            *(v4f*)(out + (size_t)row * DIM + c + 4) = h1;
        }
    }
}

// ---------------------------------------------------------------------------
extern "C" void kernel_launch(void* const* d_in, const int* in_sizes, int n_in,
                              void* d_out, int out_size, void* d_ws, size_t ws_size,
                              hipStream_t stream)
{
    const float* feat  = (const float*)d_in[0];  // [N, 64]
    const float* wfeat = (const float*)d_in[1];  // [128, 64]
    const float* w1    = (const float*)d_in[2];  // [128, 128]
    const float* w2    = (const float*)d_in[3];  // [128, 128]
    const int*   nb1   = (const int*)d_in[4];    // [N, 16]
    const int*   nb2   = (const int*)d_in[5];    // [N, 16]

    const int N = in_sizes[0] / F_IN;
    const int nblk = (N + MT - 1) / MT;

    float* x0 = (float*)d_ws;                 // [N,128] embedded features
    float* x1 = x0 + (size_t)N * DIM;         // [N,128] layer-1 output

    sage_feat_kernel<<<nblk, 256, 0, stream>>>(feat, wfeat, x0, N);
    sage_layer_kernel<<<nblk, 256, 0, stream>>>(x0, nb1, w1, x1, N);
    sage_layer_kernel<<<nblk, 256, 0, stream>>>(x1, nb2, w2, (float*)d_out, N);
}